// ProductSplineKAN_51934744543445
// MI455X (gfx1250) — compile-verified
//
#include <hip/hip_runtime.h>
#include <stdint.h>

// Problem constants (from reference setup_inputs)
#define B_N      8192
#define D_N      768
#define P_N      384
#define O_N      16
#define GRID_N   16

#define THREADS  256         // 8 wave32s
#define NCHUNK   16          // p-chunks for the split path (384/16 = 24 p each)

// coeffs slice for one p: [O=16][GRID=16][GRID=16][3] = 12288 dwords = 48 KB
#define SLICE_DW   12288
#define OBLK_DW    768        // dwords per o-block (16*16*3)
#define OBLK_PAD   772        // padded stride: 772 % 64 == 4 -> conflict-free gathers
#define SLICE_PAD  (O_N * OBLK_PAD)   // 12352 dwords = 49408 B
#define B128_PER_SLICE (SLICE_DW / 4) // 3072 16-byte transfers
#define B128_PER_THREAD (B128_PER_SLICE / THREADS) // 12

// ---- CDNA5 async global -> LDS helpers (builtin confirmed present round 1) ----

typedef int v4i __attribute__((ext_vector_type(4)));
typedef __attribute__((address_space(1))) v4i* g_v4i_ptr;  // global ptr to 128b
typedef __attribute__((address_space(3))) v4i* l_v4i_ptr;  // LDS ptr to 128b

__device__ __forceinline__ uint32_t lds_addr32(const void* p) {
  // generic pointer to __shared__ carries the LDS byte offset in its low 32 bits
  return (uint32_t)(uintptr_t)p;
}

__device__ __forceinline__ void async_copy_b128_to_lds(uint32_t lds_byte, const float* g) {
#if defined(__AMDGCN__) && __has_builtin(__builtin_amdgcn_global_load_async_to_lds_b128)
  __builtin_amdgcn_global_load_async_to_lds_b128(
      (g_v4i_ptr)(uintptr_t)g,
      (l_v4i_ptr)lds_byte,
      /*offset=*/0, /*cpol=*/0);
#else
  asm volatile("global_load_async_to_lds_b128 %0, %1, off"
               :
               : "v"(lds_byte), "v"((uint64_t)(uintptr_t)g)
               : "memory");
#endif
}

__device__ __forceinline__ void wait_asynccnt0() {
#if defined(__AMDGCN__) && __has_builtin(__builtin_amdgcn_s_wait_asynccnt)
  __builtin_amdgcn_s_wait_asynccnt(0);
#else
  asm volatile("s_wait_asynccnt 0x0" ::: "memory");
#endif
}

// Per-row staged info: one 16B record -> single ds_load_b128 broadcast per row
struct __align__(16) RowInfo {
  float a;
  float b;
  int   off;   // (ia*16+ib)*3 dword offset inside an o-block
  int   pad;
};

// ---- Main kernel (templated) ----
// TB_  : batch rows per workgroup
// PCNT : p values per workgroup (SPLIT: chunk of blockIdx.y; else all P_N)
// SPLIT: true  -> write raw partial sums to ws[blockIdx.y][b][o]
//        false -> write final out = acc + bias
template<int TB_, int PCNT, bool SPLIT>
__global__ __launch_bounds__(THREADS)
void kan_main(const float* __restrict__ x,
              const float* __restrict__ coeffs,
              const float* __restrict__ bias,
              float* __restrict__ dst) {
  constexpr int NACC = TB_ / 16;

  __shared__ float   sCoef[SLICE_PAD];   // 49408 B, layout [o][772]
  __shared__ RowInfo sRow[TB_];          // 16 B per row

  const int tid   = threadIdx.x;
  const int o     = tid & (O_N - 1);     // output index
  const int rg    = tid >> 4;            // row group 0..15
  const int bbase = blockIdx.x * TB_;
  const int pbase = SPLIT ? (int)blockIdx.y * PCNT : 0;

  float acc[NACC];
#pragma unroll
  for (int j = 0; j < NACC; ++j) acc[j] = 0.0f;

  const uint32_t sCoefBase = lds_addr32(&sCoef[0]);

  for (int i = 0; i < PCNT; ++i) {
    const int p = pbase + i;

    // ---- 1) issue async staged copy of slice p (padding inserted per o-block) ----
    const float* gsl = coeffs + (size_t)p * SLICE_DW;
#pragma unroll
    for (int k = 0; k < B128_PER_THREAD; ++k) {
      int q  = tid + k * THREADS;          // 0..3071 (b128 transfer index)
      int oo = q / 192;                    // o-block (192 b128 per o-block)
      int w  = q - oo * 192;               // b128 index inside o-block
      uint32_t ldsb = sCoefBase + (uint32_t)(oo * OBLK_PAD + w * 4) * 4u;
      async_copy_b128_to_lds(ldsb, gsl + (size_t)oo * OBLK_DW + (size_t)w * 4);
    }

    // ---- 2) per-row normalization + grid index ----
    for (int r = tid; r < TB_; r += THREADS) {
      const float* xp = x + (size_t)(bbase + r) * D_N + 2 * p;
      float xa = xp[0];
      float xb = xp[1];
      float na = fminf(fmaxf((xa + 1.0f) * 0.5f, 0.0f), 0.999999f);
      float nb = fminf(fmaxf((xb + 1.0f) * 0.5f, 0.0f), 0.999999f);
      int ia = (int)(na * (float)GRID_N);  // in [0,15] by construction
      int ib = (int)(nb * (float)GRID_N);
      RowInfo ri;
      ri.a = na; ri.b = nb; ri.off = (ia * GRID_N + ib) * 3; ri.pad = 0;
      sRow[r] = ri;                         // ds_store_b128
    }

    wait_asynccnt0();
    __syncthreads();

    // ---- 3) conflict-free LDS gathers + accumulate ----
    const float* cb = sCoef + o * OBLK_PAD;  // banks: 4*o + comp -> no conflicts
#pragma unroll
    for (int j = 0; j < NACC; ++j) {
      int r = rg + 16 * j;
      RowInfo ri = sRow[r];                  // ds_load_b128 broadcast
      float c0 = cb[ri.off + 0];
      float c1 = cb[ri.off + 1];
      float c2 = cb[ri.off + 2];
      acc[j] += fmaf(ri.a, c1, fmaf(ri.b, c2, c0));
    }

    __syncthreads();  // protect sCoef / sRow before next iteration overwrites
  }

  // ---- epilogue (deterministic, no atomics) ----
  if (SPLIT) {
    float* w = dst + (size_t)blockIdx.y * (B_N * O_N);
#pragma unroll
    for (int j = 0; j < NACC; ++j) {
      int r = rg + 16 * j;
      w[(size_t)(bbase + r) * O_N + o] = acc[j];
    }
  } else {
    const float bo = bias[o];
#pragma unroll
    for (int j = 0; j < NACC; ++j) {
      int r = rg + 16 * j;
      dst[(size_t)(bbase + r) * O_N + o] = acc[j] + bo;
    }
  }
}

// Fixed-order reduction of the NCHUNK partials + bias (deterministic)
__global__ __launch_bounds__(THREADS)
void kan_reduce(const float* __restrict__ ws,
                const float* __restrict__ bias,
                float* __restrict__ out) {
  int i = blockIdx.x * THREADS + threadIdx.x;   // 0 .. B_N*O_N-1
  if (i < B_N * O_N) {
    float s = bias[i & (O_N - 1)];
#pragma unroll
    for (int c = 0; c < NCHUNK; ++c) {
      s += ws[(size_t)c * (B_N * O_N) + i];
    }
    out[i] = s;
  }
}

extern "C" void kernel_launch(void* const* d_in, const int* in_sizes, int n_in,
                              void* d_out, int out_size, void* d_ws, size_t ws_size,
                              hipStream_t stream) {
  (void)in_sizes; (void)n_in; (void)out_size;
  const float* x      = (const float*)d_in[0];
  const float* coeffs = (const float*)d_in[1];
  const float* bias   = (const float*)d_in[2];
  float* out          = (float*)d_out;

  const size_t need = (size_t)NCHUNK * B_N * O_N * sizeof(float);  // 8 MB partials
  if (ws_size >= need && d_ws != nullptr) {
    // Split path: 32 b-tiles x 16 p-chunks = 512 WGs; 604 MB L2 coeff traffic
    kan_main<256, P_N / NCHUNK, true>
        <<<dim3(B_N / 256, NCHUNK), dim3(THREADS), 0, stream>>>(
            x, coeffs, bias, (float*)d_ws);
    kan_reduce<<<dim3((B_N * O_N) / THREADS), dim3(THREADS), 0, stream>>>(
        (const float*)d_ws, bias, out);
  } else {
    // Fallback: each WG walks all 384 p for 64 rows (no workspace needed)
    kan_main<64, P_N, false>
        <<<dim3(B_N / 64), dim3(THREADS), 0, stream>>>(
            x, coeffs, bias, out);
  }
}